// RLIFLayer_28613072126566
// MI455X (gfx1250) — compile-verified
//
#include <hip/hip_runtime.h>
#include <cstdint>
#include <cstddef>

// ---------------------------------------------------------------------------
// RLIF layer for MI455X (gfx1250), fp32 end-to-end via V_WMMA_F32_16X16X4_F32
//   Phase 0: prep    -> V0^T (diag zeroed) + clamped alpha into workspace
//   Phase 1: GEMM    -> Wx[b*T+t, h] = x[b,t,:] . W[h,:]   (memory bound)
//   Phase 2: scan    -> 4 workgroups x 16 batch rows, 1000 serial steps:
//                       spikes ping-pong in LDS (WMMA A operand),
//                       u held in WMMA C-layout registers,
//                       wx_t staged one step ahead via
//                       GLOBAL_LOAD_ASYNC_TO_LDS_B128 (ASYNCcnt),
//                       V0^T streamed from the 192MB L2 as WMMA B operand.
// ---------------------------------------------------------------------------

typedef __attribute__((ext_vector_type(2))) float v2f;
typedef __attribute__((ext_vector_type(8))) float v8f;

#define B_  64
#define T_  1000
#define IN_ 512
#define H_  512

#define ALPHA_LO 0.8187307530779818f   /* exp(-1/5)  */
#define ALPHA_HI 0.9607894391523232f   /* exp(-1/25) */

static __device__ __forceinline__ v8f wmma_f32(v2f a, v2f b, v8f c) {
  // 8-arg form: (neg_a, A, neg_b, B, c_mod, C, reuse_a, reuse_b)
  return __builtin_amdgcn_wmma_f32_16x16x4_f32(false, a, false, b,
                                               (short)0, c, false, false);
}

// Low 32 bits of a generic pointer to an LDS object ARE the wave-relative LDS
// byte offset (ISA: LDS aperture occupies the high dword).
static __device__ __forceinline__ uint32_t lds_off_u32(const void* p) {
  return (uint32_t)(uintptr_t)p;
}

// GLOBAL_LOAD_ASYNC_TO_LDS_B128: lane-wise async copy global -> LDS, ASYNCcnt.
static __device__ __forceinline__ void async_load_b128(uint32_t lds_byte_off,
                                                       const float* gaddr) {
  asm volatile("global_load_async_to_lds_b128 %0, %1, off"
               :: "v"(lds_byte_off), "v"(gaddr)
               : "memory");
}

static __device__ __forceinline__ void wait_asynccnt_le2() {
  asm volatile("s_wait_asynccnt 0x2" ::: "memory");
}

// ---------------------------------------------------------------------------
// Phase 0: V0T[h*H + j] = (j==h) ? 0 : V[j*H + h];  ac = clamp(alpha)
// ---------------------------------------------------------------------------
__global__ void prep_kernel(const float* __restrict__ V,
                            const float* __restrict__ alpha,
                            float* __restrict__ V0T,
                            float* __restrict__ ac) {
  int i = blockIdx.x * blockDim.x + threadIdx.x;
  if (i < H_ * H_) {
    int j = i / H_;            // presynaptic (row of V)
    int h = i - j * H_;        // postsynaptic (col of V)
    float v = (j == h) ? 0.0f : V[i];
    V0T[(size_t)h * H_ + j] = v;
  }
  if (i < H_) {
    ac[i] = fminf(fmaxf(alpha[i], ALPHA_LO), ALPHA_HI);
  }
}

// ---------------------------------------------------------------------------
// Phase 1: Wx = x . W^T   (M = B*T = 64000, N = H = 512, K = IN = 512)
// grid.x = M/16 blocks of 256 threads; wave w covers N columns [w*64, w*64+64)
// A fragment: lane L holds x[m0 + (L&15), k0 + 2*(L>>4) + {0,1}]   (b64 load)
// B fragment: lane L holds W[n   , k0 + 2*(L>>4) + {0,1}]          (b64 load)
// ---------------------------------------------------------------------------
__global__ void __launch_bounds__(256) wx_gemm_kernel(
    const float* __restrict__ x, const float* __restrict__ W,
    float* __restrict__ wx) {
  const int lane = threadIdx.x & 31;
  const int wave = threadIdx.x >> 5;       // 0..7
  const int m0   = blockIdx.x * 16;
  const int n0   = wave * 64;
  const int row  = lane & 15;
  const int kg   = (lane >> 4) << 1;       // 0 or 2

  const float* xrow = x + (size_t)(m0 + row) * IN_ + kg;
  const float* w0   = W + (size_t)(n0 +  0 + row) * IN_ + kg;
  const float* w1   = W + (size_t)(n0 + 16 + row) * IN_ + kg;
  const float* w2   = W + (size_t)(n0 + 32 + row) * IN_ + kg;
  const float* w3   = W + (size_t)(n0 + 48 + row) * IN_ + kg;

  v8f acc0 = {}, acc1 = {}, acc2 = {}, acc3 = {};
#pragma unroll 4
  for (int k0 = 0; k0 < IN_; k0 += 4) {
    v2f a  = *(const v2f*)(xrow + k0);
    v2f b0 = *(const v2f*)(w0 + k0);
    v2f b1 = *(const v2f*)(w1 + k0);
    v2f b2 = *(const v2f*)(w2 + k0);
    v2f b3 = *(const v2f*)(w3 + k0);
    acc0 = wmma_f32(a, b0, acc0);
    acc1 = wmma_f32(a, b1, acc1);
    acc2 = wmma_f32(a, b2, acc2);
    acc3 = wmma_f32(a, b3, acc3);
  }

  // C layout: vgpr r, lane L -> (M = r + 8*(L>>4), N = tile*16 + (L&15))
  const int mrow = m0 + (lane >> 4) * 8;
  float* o = wx + (size_t)mrow * H_ + n0 + row;
#pragma unroll
  for (int r = 0; r < 8; ++r) {
    o[(size_t)r * H_ +  0] = acc0[r];
    o[(size_t)r * H_ + 16] = acc1[r];
    o[(size_t)r * H_ + 32] = acc2[r];
    o[(size_t)r * H_ + 48] = acc3[r];
  }
}

// ---------------------------------------------------------------------------
// Phase 2: recurrent scan. 4 blocks x 1024 threads (32 waves).
// Block owns 16 batch rows; wave owns one 16-column slice of H.
// Each wave stages the wx_t slice it will consume one step ahead with async
// global->LDS b128 copies (2 per lane per step, so s_wait_asynccnt<=2 makes
// the current buffer resident) -- no cross-wave dependency, no extra barrier.
// ---------------------------------------------------------------------------
__global__ void __launch_bounds__(1024) rlif_scan_kernel(
    const float* __restrict__ wx,    // [B*T, H]  (m = b*T + t)
    const float* __restrict__ V0T,   // [H, H]    V0 transposed, diag zeroed
    const float* __restrict__ ac,    // [H]       clamped alpha
    const float* __restrict__ ut0,   // [B, H]
    const float* __restrict__ st0,   // [B, H]
    float* __restrict__ out)         // [B, T, H]
{
  __shared__ float S[2][16][H_];          // 64 KB spike ping-pong (WMMA A)
  __shared__ float Swx[2][32][16][16];    // 64 KB staged wx tiles, per wave

  const int lane = threadIdx.x & 31;
  const int wave = threadIdx.x >> 5;     // 0..31
  const int b0   = blockIdx.x * 16;
  const int row  = lane & 15;
  const int hi   = lane >> 4;            // 0/1 lane half
  const int kg   = hi << 1;              // K sub-offset 0 or 2
  const int h    = wave * 16 + row;      // this lane's output column

  const float a   = ac[h];
  const float oma = 1.0f - a;

  // Membrane potential in WMMA C-layout registers.
  v8f u;
#pragma unroll
  for (int r = 0; r < 8; ++r)
    u[r] = ut0[(size_t)(b0 + hi * 8 + r) * H_ + h];

  // Initial spikes into LDS buffer 0.
  for (int i = threadIdx.x; i < 16 * H_; i += 1024)
    S[0][i >> 9][i & (H_ - 1)] = st0[(size_t)(b0 + (i >> 9)) * H_ + (i & (H_ - 1))];
  __syncthreads();

  // Async staging pattern: wave's wx tile is 16 rows(b) x 16 cols(h) = 1 KB
  // = 64 x 16B chunks; lane handles chunks {lane, lane+32}.
  const int bl0 = lane >> 2,        co0 = (lane & 3) << 2;
  const int bl1 = (lane + 32) >> 2, co1 = ((lane + 32) & 3) << 2;
  const float* g0 = wx + (size_t)(b0 + bl0) * T_ * H_ + wave * 16 + co0;
  const float* g1 = wx + (size_t)(b0 + bl1) * T_ * H_ + wave * 16 + co1;
  uint32_t l0[2], l1[2];
#pragma unroll
  for (int bf = 0; bf < 2; ++bf) {
    l0[bf] = lds_off_u32(&Swx[bf][wave][bl0][co0]);
    l1[bf] = lds_off_u32(&Swx[bf][wave][bl1][co1]);
  }

  // Prologue: stage step 0 into buffer 0.
  async_load_b128(l0[0], g0);
  async_load_b128(l1[0], g1);

  const float* vrow = V0T + (size_t)h * H_ + kg;   // B fragments, b64 loads

  int cur = 0;
  for (int t = 0; t < T_; ++t) {
    const int cbuf = t & 1;
    const int nbuf = cbuf ^ 1;

    // Kick off next step's wx tile (benign re-load of the last step at t=T-1
    // keeps the per-step async count uniform at 2).
    const int t1 = (t + 1 < T_) ? (t + 1) : t;
    async_load_b128(l0[nbuf], g0 + (size_t)t1 * H_);
    async_load_b128(l1[nbuf], g1 + (size_t)t1 * H_);

    // z = S @ V0 for this wave's 16 columns; 4 accumulator chains over K=512.
    v8f z0 = {}, z1 = {}, z2 = {}, z3 = {};
    const float* srow = &S[cur][row][kg];          // A fragments, ds b64 loads
#pragma unroll 2
    for (int k0 = 0; k0 < H_; k0 += 16) {
      v2f a0 = *(const v2f*)(srow + k0);
      v2f a1 = *(const v2f*)(srow + k0 + 4);
      v2f a2 = *(const v2f*)(srow + k0 + 8);
      v2f a3 = *(const v2f*)(srow + k0 + 12);
      v2f b0 = *(const v2f*)(vrow + k0);
      v2f b1 = *(const v2f*)(vrow + k0 + 4);
      v2f b2 = *(const v2f*)(vrow + k0 + 8);
      v2f b3 = *(const v2f*)(vrow + k0 + 12);
      z0 = wmma_f32(a0, b0, z0);
      z1 = wmma_f32(a1, b1, z1);
      z2 = wmma_f32(a2, b2, z2);
      z3 = wmma_f32(a3, b3, z3);
    }

    // Current step's staged wx tile is resident once ASYNCcnt <= 2
    // (this wave's own 2 outstanding loads are for step t+1).
    wait_asynccnt_le2();

    // LIF update directly on the C-layout fragments.
    const int nxt = cur ^ 1;
#pragma unroll
    for (int r = 0; r < 8; ++r) {
      const int   bl    = hi * 8 + r;
      const float wxv   = Swx[cbuf][wave][bl][row];
      const float sprev = S[cur][bl][h];
      const float z     = z0[r] + z1[r] + z2[r] + z3[r];
      const float un    = a * (u[r] - sprev) + oma * (wxv + z);
      u[r] = un;
      const float sn = (un > 1.0f) ? 1.0f : 0.0f;      // spike(u - 1)
      S[nxt][bl][h] = sn;
      out[((size_t)(b0 + bl) * T_ + t) * H_ + h] = sn;
    }
    __syncthreads();
    cur = nxt;
  }
}

// ---------------------------------------------------------------------------
// Host launcher
// ---------------------------------------------------------------------------
extern "C" void kernel_launch(void* const* d_in, const int* in_sizes, int n_in,
                              void* d_out, int out_size, void* d_ws, size_t ws_size,
                              hipStream_t stream) {
  const float* x     = (const float*)d_in[0];   // [B,T,IN]
  const float* W     = (const float*)d_in[1];   // [H,IN]
  const float* V     = (const float*)d_in[2];   // [H,H]
  const float* alpha = (const float*)d_in[3];   // [H]
  const float* ut0   = (const float*)d_in[4];   // [B,H]
  const float* st0   = (const float*)d_in[5];   // [B,H]
  // d_in[6] = input_layer (unused: substeps == 1)
  float* out = (float*)d_out;                   // [B,T,H]

  const size_t WX_BYTES = (size_t)B_ * T_ * H_ * sizeof(float);   // 131 MB
  const size_t V0_BYTES = (size_t)H_ * H_ * sizeof(float);        // 1 MB

  char*  ws  = (char*)d_ws;
  float* Wx  = (float*)(ws);
  float* V0T = (float*)(ws + WX_BYTES);
  float* ac  = (float*)(ws + WX_BYTES + V0_BYTES);

  prep_kernel<<<(H_ * H_ + 255) / 256, 256, 0, stream>>>(V, alpha, V0T, ac);
  wx_gemm_kernel<<<(B_ * T_) / 16, 256, 0, stream>>>(x, W, Wx);
  rlif_scan_kernel<<<B_ / 16, 1024, 0, stream>>>(Wx, V0T, ac, ut0, st0, out);
}